// DM_DDI_26087631356312
// MI455X (gfx1250) — compile-verified
//
#include <hip/hip_runtime.h>
#include <hip/hip_bf16.h>
#include <math.h>

// ---------------------------------------------------------------------------
// Problem constants (match reference)
// ---------------------------------------------------------------------------
#define NN      10000
#define EE      100000
#define PP      200000
#define NT_     160000
#define N_IN    1716
#define N_E1    2000
#define N_E2    256
#define N_ZZ    128
#define ATT_HH  128

// d_out flat offsets (float elements), reference return order
#define EMB1_OFF 0LL
#define BETA_OFF 1280000LL
#define XBAR_OFF 1300000LL
#define C1_OFF   18460000LL
#define C2_OFF   38940000LL
#define L1_OFF   44060000LL
#define L2_OFF   44220000LL

typedef float v2f __attribute__((ext_vector_type(2)));
typedef float v8f __attribute__((ext_vector_type(8)));
typedef unsigned u32x4 __attribute__((ext_vector_type(4)));
typedef int      i32x4 __attribute__((ext_vector_type(4)));
typedef int      i32x8 __attribute__((ext_vector_type(8)));

#if __has_builtin(__builtin_amdgcn_tensor_load_to_lds)
#define HAVE_TDM 1
#else
#define HAVE_TDM 0
#endif

#if __has_builtin(__builtin_amdgcn_s_wait_tensorcnt)
#define WAIT_TENSORCNT(n) __builtin_amdgcn_s_wait_tensorcnt((short)(n))
#else
#define WAIT_TENSORCNT(n) asm volatile("s_wait_tensorcnt %0" ::"i"(n))
#endif

// ---------------------------------------------------------------------------
// GEMM tiling: block 64x64, BK=32; 8 waves (wave32), wave grid 4(M) x 2(N),
// each wave owns 16 rows x 32 cols = two v8f accumulators.
// LDS layout (matches TDM pad-on-load): A rows stride 34 dwords (pad 2 per 32),
// B rows stride 72 dwords (pad 8 per 64) -> conflict-free fragment loads.
// ---------------------------------------------------------------------------
#define BM 64
#define BN 64
#define BK 32
#define LDA_S (BK + 2)   // 34
#define LDB_S (BN + 8)   // 72

#if HAVE_TDM
// Generic -> LDS(AS3) byte offset for TDM descriptors
#define LDS_OFF(p) ((unsigned)(unsigned long long)(__attribute__((address_space(3))) const void*)(const void*)(p))

// Issue a 2D TDM tile load: tile (tile1 rows x tile0 cols) of fp32 from a
// row-major tensor. rem0/rem1 are the remaining tensor extents measured from
// the tile origin (TDM zero-fills reads past them). pad-on-load controls the
// LDS row stride.
__device__ __forceinline__ void tdm_load_2d(const float* gptr, unsigned lds_off,
                                            int rem0, int rem1, int stride0,
                                            int tile0, int tile1,
                                            int pad_interval, int pad_amount)
{
    const unsigned long long ga = (unsigned long long)gptr;
    u32x4 g0;
    g0.x = 1u;                                        // count=1, user mode
    g0.y = lds_off;                                   // lds_addr
    g0.z = (unsigned)(ga & 0xFFFFFFFFull);            // global_addr[31:0]
    g0.w = (unsigned)((ga >> 32) & 0x01FFFFFFull)     // global_addr[56:32]
         | 0x80000000u;                               // type=2 ("image")
    const unsigned td0 = (unsigned)rem0;
    const unsigned td1 = (unsigned)rem1;
    i32x8 g1;
    g1[0] = (int)((2u << 16)                          // data_size = 4B
                | (1u << 20)                          // pad_enable
                | ((unsigned)pad_interval << 22)
                | ((unsigned)pad_amount << 25));
    g1[1] = (int)((td0 & 0xFFFFu) << 16);             // tensor_dim0[15:0]
    g1[2] = (int)(((td0 >> 16) & 0xFFFFu) | ((td1 & 0xFFFFu) << 16));
    g1[3] = (int)(((td1 >> 16) & 0xFFFFu) | ((unsigned)tile0 << 16));
    g1[4] = (int)(tile1 & 0xFFFF);                    // tile_dim1 (tile_dim2=0)
    g1[5] = stride0;                                  // tensor_dim0_stride
    g1[6] = 0;
    g1[7] = 0;
    const i32x4 z4 = {0, 0, 0, 0};
    const i32x8 z8 = {0, 0, 0, 0, 0, 0, 0, 0};
    __builtin_amdgcn_tensor_load_to_lds(g0, g1, z4, z4, z8, 0);
}
#endif

__global__ __launch_bounds__(256)
void gemm_wmma_f32(const float* __restrict__ A, const float* __restrict__ B,
                   const float* __restrict__ bias, float* __restrict__ C,
                   int M, int N, int K, int do_relu)
{
    __shared__ float sA[2][BM * LDA_S];
    __shared__ float sB[2][BK * LDB_S];

    const int tid  = threadIdx.x;
    const int lane = tid & 31;
    const int wave = tid >> 5;
    const int wm   = wave & 3;   // M sub-tile (16 rows each)
    const int wn   = wave >> 2;  // N sub-tile (32 cols each)

    const int row0 = blockIdx.y * BM;
    const int col0 = blockIdx.x * BN;

    const int lm   = lane & 15;
    const int ksel = (lane >> 4) << 1;   // lanes 16-31 hold K=2,3 (ISA 7.12.2)

    v8f acc0 = {};
    v8f acc1 = {};

    const int S = (K + BK - 1) / BK;

#if HAVE_TDM
    if (wave == 0 && lane == 0) {
        // prologue: stage 0 into buffer 0
        tdm_load_2d(A + (size_t)row0 * K, LDS_OFF(&sA[0][0]),
                    K, M - row0, K, BK, BM, /*pint 32dw*/4, /*pamt 2dw*/1);
        tdm_load_2d(B + col0, LDS_OFF(&sB[0][0]),
                    N - col0, K, N, BN, BK, /*pint 64dw*/5, /*pamt 8dw*/7);
    }
#endif

    for (int s = 0; s < S; ++s) {
#if HAVE_TDM
        const int buf = s & 1;
        if (wave == 0 && lane == 0) {
            if (s + 1 < S) {
                const int k1 = (s + 1) * BK;
                const int nxt = buf ^ 1;
                tdm_load_2d(A + (size_t)row0 * K + k1, LDS_OFF(&sA[nxt][0]),
                            K - k1, M - row0, K, BK, BM, 4, 1);
                tdm_load_2d(B + (size_t)k1 * N + col0, LDS_OFF(&sB[nxt][0]),
                            N - col0, K - k1, N, BN, BK, 5, 7);
                WAIT_TENSORCNT(2);   // stage-s pair complete (TDM is in-order)
            } else {
                WAIT_TENSORCNT(0);
            }
        }
        __syncthreads();             // release all waves onto stage-s data
#else
        const int buf = 0;
        const int k0 = s * BK;
        // stage A tile (64x32), zero-pad out-of-range
#pragma unroll
        for (int i = 0; i < 8; ++i) {
            int idx = tid + i * 256;
            int m   = idx >> 5;
            int kk  = idx & 31;
            int gr  = row0 + m, gk = k0 + kk;
            sA[0][m * LDA_S + kk] =
                (gr < M && gk < K) ? A[(size_t)gr * K + gk] : 0.0f;
        }
        // stage B tile (32x64), zero-pad out-of-range
#pragma unroll
        for (int i = 0; i < 8; ++i) {
            int idx = tid + i * 256;
            int kk  = idx >> 6;
            int n   = idx & 63;
            int gk  = k0 + kk, gc = col0 + n;
            sB[0][kk * LDB_S + n] =
                (gk < K && gc < N) ? B[(size_t)gk * N + gc] : 0.0f;
        }
        __syncthreads();
#endif

        const float* As = &sA[buf][0];
        const float* Bs = &sB[buf][0];
#pragma unroll
        for (int kk = 0; kk < BK; kk += 4) {
            const int ks = kk + ksel;
            v2f a, b0, b1;
            const float* ap = &As[(wm * 16 + lm) * LDA_S + ks];
            a.x = ap[0];
            a.y = ap[1];
            const float* bp = &Bs[ks * LDB_S + wn * 32 + lm];
            b0.x = bp[0];     b0.y = bp[LDB_S];
            b1.x = bp[16];    b1.y = bp[LDB_S + 16];
            acc0 = __builtin_amdgcn_wmma_f32_16x16x4_f32(
                       false, a, false, b0, (short)0, acc0, false, false);
            acc1 = __builtin_amdgcn_wmma_f32_16x16x4_f32(
                       false, a, false, b1, (short)0, acc1, false, false);
        }
        __syncthreads();             // protect buf before it is refilled
    }

    // epilogue: VGPR j holds rows j (lanes 0-15) / j+8 (lanes 16-31)
    const int rbase = row0 + wm * 16 + ((lane >> 4) << 3);
    const int cb0   = col0 + wn * 32 + lm;
    const int cb1   = cb0 + 16;
#pragma unroll
    for (int j = 0; j < 8; ++j) {
        int r = rbase + j;
        if (r < M) {
            if (cb0 < N) {
                float v = acc0[j] + (bias ? bias[cb0] : 0.0f);
                if (do_relu) v = fmaxf(v, 0.0f);
                C[(size_t)r * N + cb0] = v;
            }
            if (cb1 < N) {
                float v = acc1[j] + (bias ? bias[cb1] : 0.0f);
                if (do_relu) v = fmaxf(v, 0.0f);
                C[(size_t)r * N + cb1] = v;
            }
        }
    }
}

// ---------------------------------------------------------------------------
// CSR row_start from sorted adj_rows (binary search lower_bound per row)
// ---------------------------------------------------------------------------
__global__ void row_start_kernel(const int* __restrict__ rows,
                                 int* __restrict__ row_start, int n, int e)
{
    int r = blockIdx.x * blockDim.x + threadIdx.x;
    if (r > n) return;
    int lo = 0, hi = e;
    while (lo < hi) {
        int mid = (lo + hi) >> 1;
        if (rows[mid] < r) lo = mid + 1; else hi = mid;
    }
    row_start[r] = lo;
}

// ---------------------------------------------------------------------------
// Deterministic SpMM + ReLU: out[r,:] = relu( sum_e vals[e]*in[cols[e],:] )
// One (row, 256-feature chunk) per block; in-order accumulation per thread.
// ---------------------------------------------------------------------------
__global__ __launch_bounds__(256)
void spmm_relu_kernel(const int* __restrict__ row_start,
                      const int* __restrict__ cols,
                      const float* __restrict__ vals,
                      const float* __restrict__ in,
                      float* __restrict__ out, int F)
{
    int f = blockIdx.x * blockDim.x + threadIdx.x;
    int r = blockIdx.y;
    if (f >= F) return;
    float acc = 0.0f;
    const int e0 = row_start[r], e1 = row_start[r + 1];
    for (int e = e0; e < e1; ++e)
        acc = fmaf(vals[e], in[(size_t)cols[e] * F + f], acc);
    out[(size_t)r * F + f] = fmaxf(acc, 0.0f);
}

// ---------------------------------------------------------------------------
// mix = (1-sigma)*h + sigma*t   (sigma = 0.5)
// ---------------------------------------------------------------------------
__global__ __launch_bounds__(256)
void mix_kernel(const float* __restrict__ h, const float* __restrict__ t,
                float* __restrict__ out, long n)
{
    long i = (long)blockIdx.x * blockDim.x + threadIdx.x;
    if (i < n) out[i] = 0.5f * (h[i] + t[i]);
}

// ---------------------------------------------------------------------------
// Attention fusion: per node, w_k = tanh(e_k @ Wa1 + ba1) @ Wa2 for
// e_0 = h3[i], e_1 = z[i]; beta = softmax(w); emb1 = beta0*e0 + beta1*e1.
// ---------------------------------------------------------------------------
__global__ __launch_bounds__(128)
void attention_kernel(const float* __restrict__ h3, const float* __restrict__ z,
                      const float* __restrict__ Wa1, const float* __restrict__ ba1,
                      const float* __restrict__ Wa2,
                      float* __restrict__ emb1_out, float* __restrict__ beta_out)
{
    __shared__ float e0[N_ZZ], e1[N_ZZ], redA[ATT_HH], redB[ATT_HH];
    const int j = threadIdx.x;
    const int i = blockIdx.x;

    e0[j] = h3[(size_t)i * N_ZZ + j];
    e1[j] = z [(size_t)i * N_ZZ + j];
    __syncthreads();

    float t0 = 0.0f, t1 = 0.0f;
#pragma unroll 8
    for (int d = 0; d < N_ZZ; ++d) {
        const float w = Wa1[d * ATT_HH + j];
        t0 = fmaf(e0[d], w, t0);
        t1 = fmaf(e1[d], w, t1);
    }
    const float wa2 = Wa2[j];
    redA[j] = tanhf(t0 + ba1[j]) * wa2;
    redB[j] = tanhf(t1 + ba1[j]) * wa2;
    __syncthreads();
    for (int off = ATT_HH / 2; off > 0; off >>= 1) {
        if (j < off) { redA[j] += redA[j + off]; redB[j] += redB[j + off]; }
        __syncthreads();
    }
    const float w0 = redA[0], w1 = redB[0];
    const float m  = fmaxf(w0, w1);
    const float x0 = __expf(w0 - m), x1 = __expf(w1 - m);
    const float inv = 1.0f / (x0 + x1);
    const float b0 = x0 * inv, b1 = x1 * inv;

    emb1_out[(size_t)i * N_ZZ + j] = b0 * e0[j] + b1 * e1[j];
    if (j == 0) { beta_out[i * 2 + 0] = b0; beta_out[i * 2 + 1] = b1; }
}

// ---------------------------------------------------------------------------
// Pair gather: C = 0.5*(emb1[i0] + emb1[i1]); split into C1 (first NT) / C2.
// ---------------------------------------------------------------------------
__global__ __launch_bounds__(256)
void pairs_kernel(const float* __restrict__ emb1, const int* __restrict__ pair_idx,
                  float* __restrict__ out)
{
    long idx = (long)blockIdx.x * blockDim.x + threadIdx.x;
    long p = idx >> 7;          // /128
    int  f = (int)(idx & 127);
    if (p >= PP) return;
    const int i0 = pair_idx[p * 2 + 0];
    const int i1 = pair_idx[p * 2 + 1];
    const float v = 0.5f * (emb1[(size_t)i0 * N_ZZ + f] + emb1[(size_t)i1 * N_ZZ + f]);
    if (p < NT_) out[C1_OFF + p * N_ZZ + f] = v;
    else         out[C2_OFF + (p - NT_) * N_ZZ + f] = v;
}

__global__ __launch_bounds__(256)
void labels_kernel(const int* __restrict__ labels, float* __restrict__ out)
{
    long p = (long)blockIdx.x * blockDim.x + threadIdx.x;
    if (p >= PP) return;
    const float v = (float)labels[p];
    if (p < NT_) out[L1_OFF + p] = v;
    else         out[L2_OFF + (p - NT_)] = v;
}

// ---------------------------------------------------------------------------
// Launch
// ---------------------------------------------------------------------------
static inline dim3 gemm_grid(int M, int N) {
    return dim3((N + BN - 1) / BN, (M + BM - 1) / BM);
}

extern "C" void kernel_launch(void* const* d_in, const int* in_sizes, int n_in,
                              void* d_out, int out_size, void* d_ws, size_t ws_size,
                              hipStream_t stream)
{
    (void)in_sizes; (void)n_in; (void)out_size; (void)ws_size;
    const float* x        = (const float*)d_in[0];
    const int*   adj_rows = (const int*)  d_in[1];
    const int*   adj_cols = (const int*)  d_in[2];
    const float* adj_vals = (const float*)d_in[3];
    const int*   pair_idx = (const int*)  d_in[4];
    const int*   labels   = (const int*)  d_in[5];
    const float* W_enc1 = (const float*)d_in[6];  const float* b_enc1 = (const float*)d_in[7];
    const float* W_enc2 = (const float*)d_in[8];  const float* b_enc2 = (const float*)d_in[9];
    const float* W_z    = (const float*)d_in[10]; const float* b_z    = (const float*)d_in[11];
    const float* W_dec1 = (const float*)d_in[12]; const float* b_dec1 = (const float*)d_in[13];
    const float* W_dec2 = (const float*)d_in[14]; const float* b_dec2 = (const float*)d_in[15];
    const float* W_xbar = (const float*)d_in[16]; const float* b_xbar = (const float*)d_in[17];
    const float* W_g1   = (const float*)d_in[18];
    const float* W_g2   = (const float*)d_in[19];
    const float* W_g3   = (const float*)d_in[20];
    const float* W_a1   = (const float*)d_in[21]; const float* b_a1   = (const float*)d_in[22];
    const float* W_a2   = (const float*)d_in[23];

    float* dout = (float*)d_out;

    // workspace carve-up (floats); dec2 buffer is recycled for the GCN branch
    float* w = (float*)d_ws;
    float* tra1 = w; w += (size_t)NN * N_E1;   // 20M
    float* tra2 = w; w += (size_t)NN * N_E2;   // 2.56M
    float* zb   = w; w += (size_t)NN * N_ZZ;   // 1.28M
    float* dec1 = w; w += (size_t)NN * N_E2;   // 2.56M
    float* dec2 = w; w += (size_t)NN * N_E1;   // 20M  (reused: xg1 / mix1)
    float* h1   = w; w += (size_t)NN * N_E1;   // 20M
    float* t2   = w; w += (size_t)NN * N_E2;   // 2.56M (reused: mix2)
    float* h2   = w; w += (size_t)NN * N_E2;   // 2.56M
    float* t3   = w; w += (size_t)NN * N_ZZ;   // 1.28M
    float* h3   = w; w += (size_t)NN * N_ZZ;   // 1.28M
    int* row_start = (int*)w;                  // N+1 ints

    const dim3 blk(256);

    // ---- autoencoder ----
    gemm_wmma_f32<<<gemm_grid(NN, N_E1), blk, 0, stream>>>(x,    W_enc1, b_enc1, tra1, NN, N_E1, N_IN, 1);
    gemm_wmma_f32<<<gemm_grid(NN, N_E2), blk, 0, stream>>>(tra1, W_enc2, b_enc2, tra2, NN, N_E2, N_E1, 1);
    gemm_wmma_f32<<<gemm_grid(NN, N_ZZ), blk, 0, stream>>>(tra2, W_z,    b_z,    zb,   NN, N_ZZ, N_E2, 0);
    gemm_wmma_f32<<<gemm_grid(NN, N_E2), blk, 0, stream>>>(zb,   W_dec1, b_dec1, dec1, NN, N_E2, N_ZZ, 1);
    gemm_wmma_f32<<<gemm_grid(NN, N_E1), blk, 0, stream>>>(dec1, W_dec2, b_dec2, dec2, NN, N_E1, N_E2, 1);
    gemm_wmma_f32<<<gemm_grid(NN, N_IN), blk, 0, stream>>>(dec2, W_xbar, b_xbar, dout + XBAR_OFF, NN, N_IN, N_E1, 0);

    // ---- GCN branch (dec2 buffer now free -> xg1/mix1) ----
    row_start_kernel<<<(NN + 1 + 255) / 256, blk, 0, stream>>>(adj_rows, row_start, NN, EE);

    gemm_wmma_f32<<<gemm_grid(NN, N_E1), blk, 0, stream>>>(x, W_g1, nullptr, dec2, NN, N_E1, N_IN, 0);
    spmm_relu_kernel<<<dim3((N_E1 + 255) / 256, NN), blk, 0, stream>>>(row_start, adj_cols, adj_vals, dec2, h1, N_E1);
    mix_kernel<<<(int)(((long)NN * N_E1 + 255) / 256), blk, 0, stream>>>(h1, tra1, dec2, (long)NN * N_E1);

    gemm_wmma_f32<<<gemm_grid(NN, N_E2), blk, 0, stream>>>(dec2, W_g2, nullptr, t2, NN, N_E2, N_E1, 0);
    spmm_relu_kernel<<<dim3((N_E2 + 255) / 256, NN), blk, 0, stream>>>(row_start, adj_cols, adj_vals, t2, h2, N_E2);
    mix_kernel<<<(int)(((long)NN * N_E2 + 255) / 256), blk, 0, stream>>>(h2, tra2, t2, (long)NN * N_E2);

    gemm_wmma_f32<<<gemm_grid(NN, N_ZZ), blk, 0, stream>>>(t2, W_g3, nullptr, t3, NN, N_ZZ, N_E2, 0);
    spmm_relu_kernel<<<dim3((N_ZZ + 255) / 256, NN), blk, 0, stream>>>(row_start, adj_cols, adj_vals, t3, h3, N_ZZ);

    // ---- attention fusion -> emb1, beta (written directly into d_out) ----
    attention_kernel<<<NN, 128, 0, stream>>>(h3, zb, W_a1, b_a1, W_a2,
                                             dout + EMB1_OFF, dout + BETA_OFF);

    // ---- pair gather + labels ----
    pairs_kernel<<<(int)(((long)PP * N_ZZ + 255) / 256), blk, 0, stream>>>(dout + EMB1_OFF, pair_idx, dout);
    labels_kernel<<<(PP + 255) / 256, blk, 0, stream>>>(labels, dout);
}